// P2PFraudGNN_39977555591795
// MI455X (gfx1250) — compile-verified
//
#include <hip/hip_runtime.h>
#include <hip/hip_bf16.h>

constexpr int   kN    = 50000;
constexpr int   kE    = 600000;
constexpr int   kH    = 128;     // hidden dim (also F)
constexpr int   kM    = 200000;
constexpr int   kTotE = kE + kN; // edges + self loops
constexpr float kEps  = 1e-5f;
constexpr float kSlope = 0.2f;

typedef __attribute__((ext_vector_type(2))) float v2f;
typedef __attribute__((ext_vector_type(8))) float v8f;

// ------------------------------------------------------------------
// utility kernels
// ------------------------------------------------------------------
__global__ void fill_kernel(float* __restrict__ p, float v, int n) {
  int i = blockIdx.x * blockDim.x + threadIdx.x;
  if (i < n) p[i] = v;
}

__device__ __forceinline__ void atomic_max_f32(float* addr, float val) {
  unsigned int* ua = reinterpret_cast<unsigned int*>(addr);
  unsigned int cur = *ua;
  while (__uint_as_float(cur) < val) {
    unsigned int assumed = cur;
    cur = atomicCAS(ua, assumed, __float_as_uint(val));
    if (cur == assumed) break;
  }
}

// ------------------------------------------------------------------
// fp32 WMMA GEMM: Out[n x OC] = A[n x K] @ W[K x OC], all row-major.
// One wave per 16x16 output tile; K multiple of 4; n, OC multiples of 16.
// A frag (16x4):  lane l: v0=A[l&15][k+2*(l>>4)], v1=A[l&15][k+2*(l>>4)+1]
// B frag (4x16):  lane l: v0=B[k+2*(l>>4)][l&15], v1=B[k+2*(l>>4)+1][l&15]
// D (16x16):      lane l, reg r: D[r + 8*(l>>4)][l&15]
// ------------------------------------------------------------------
__global__ void __launch_bounds__(32) gemm_wmma_f32_kernel(
    const float* __restrict__ A, const float* __restrict__ W,
    float* __restrict__ Out, int K, int OC) {
  const int r0 = blockIdx.x * 16;
  const int c0 = blockIdx.y * 16;
  const int l  = threadIdx.x;
  const int lr = l & 15;
  const int hi = l >> 4;
  const float* arow = A + (size_t)(r0 + lr) * K;
  v8f acc = {};
  for (int k = 0; k < K; k += 4) {
    const int ka = k + 2 * hi;
    v2f a, b;
    a.x = arow[ka];
    a.y = arow[ka + 1];
    b.x = W[(size_t)ka * OC + c0 + lr];
    b.y = W[(size_t)(ka + 1) * OC + c0 + lr];
    acc = __builtin_amdgcn_wmma_f32_16x16x4_f32(false, a, false, b, (short)0, acc,
                                                false, false);
  }
#pragma unroll
  for (int r = 0; r < 8; ++r)
    Out[(size_t)(r0 + r + 8 * hi) * OC + c0 + lr] = acc[r];
}

// ------------------------------------------------------------------
// GAT attention pieces
// ------------------------------------------------------------------
__global__ void alpha_kernel(const float* __restrict__ h, const float* __restrict__ avec,
                             float* __restrict__ alpha, int heads, int ch) {
  int idx = blockIdx.x * blockDim.x + threadIdx.x;
  if (idx >= kN * heads) return;
  int node = idx / heads;
  int hd   = idx - node * heads;
  const float* hp = h + (size_t)node * kH + hd * ch;
  const float* ap = avec + hd * ch;
  float s = 0.f;
  for (int c = 0; c < ch; ++c) s = fmaf(hp[c], ap[c], s);
  alpha[idx] = s;
}

__global__ void att_max_kernel(const float* __restrict__ as_, const float* __restrict__ ad_,
                               const int* __restrict__ src, const int* __restrict__ dst,
                               int heads, float* __restrict__ emax) {
  int idx = blockIdx.x * blockDim.x + threadIdx.x;
  if (idx >= kTotE * heads) return;
  int e  = idx / heads;
  int hd = idx - e * heads;
  int s = (e < kE) ? src[e] : (e - kE);
  int d = (e < kE) ? dst[e] : (e - kE);
  float v = as_[s * heads + hd] + ad_[d * heads + hd];
  v = (v > 0.f) ? v : kSlope * v;
  atomic_max_f32(&emax[d * heads + hd], v);
}

__global__ void att_sum_kernel(const float* __restrict__ as_, const float* __restrict__ ad_,
                               const int* __restrict__ src, const int* __restrict__ dst,
                               int heads, const float* __restrict__ emax,
                               float* __restrict__ denom, float* __restrict__ attn) {
  int idx = blockIdx.x * blockDim.x + threadIdx.x;
  if (idx >= kTotE * heads) return;
  int e  = idx / heads;
  int hd = idx - e * heads;
  int s = (e < kE) ? src[e] : (e - kE);
  int d = (e < kE) ? dst[e] : (e - kE);
  float v = as_[s * heads + hd] + ad_[d * heads + hd];
  v = (v > 0.f) ? v : kSlope * v;
  float ee = __expf(v - emax[d * heads + hd]);
  attn[idx] = ee;
  atomicAdd(&denom[d * heads + hd], ee);
}

__global__ void att_agg_kernel(const float* __restrict__ h, const float* __restrict__ attn,
                               const float* __restrict__ denom,
                               const int* __restrict__ src, const int* __restrict__ dst,
                               int heads, int ch, float* __restrict__ agg) {
  int idx = blockIdx.x * blockDim.x + threadIdx.x;   // (kTotE*128) = 83.2M < 2^31
  if (idx >= kTotE * kH) return;
  int e = idx >> 7;
  int f = idx & 127;
  int hd = f / ch;
  int s = (e < kE) ? src[e] : (e - kE);
  int d = (e < kE) ? dst[e] : (e - kE);
  float a = attn[e * heads + hd] / (denom[d * heads + hd] + 1e-16f);
  atomicAdd(&agg[(size_t)d * kH + f], h[(size_t)s * kH + f] * a);
}

// x = bn(elu(agg + bias))
__global__ void bn_elu_kernel(const float* __restrict__ agg, const float* __restrict__ bias,
                              const float* __restrict__ g, const float* __restrict__ b,
                              float* __restrict__ xout) {
  int idx = blockIdx.x * blockDim.x + threadIdx.x;
  if (idx >= kN * kH) return;
  int col = idx & 127;
  float v = agg[idx] + bias[col];
  v = (v > 0.f) ? v : (__expf(v) - 1.f);
  xout[idx] = v * (g[col] * rsqrtf(1.f + kEps)) + b[col];
}

// ------------------------------------------------------------------
// GCN pieces
// ------------------------------------------------------------------
__global__ void deg_kernel(const int* __restrict__ dst, float* __restrict__ deg) {
  int e = blockIdx.x * blockDim.x + threadIdx.x;
  if (e >= kTotE) return;
  int d = (e < kE) ? dst[e] : (e - kE);
  atomicAdd(&deg[d], 1.0f);
}

__global__ void gcn_agg_kernel(const float* __restrict__ h, const float* __restrict__ deg,
                               const int* __restrict__ src, const int* __restrict__ dst,
                               float* __restrict__ agg) {
  int idx = blockIdx.x * blockDim.x + threadIdx.x;
  if (idx >= kTotE * kH) return;
  int e = idx >> 7;
  int f = idx & 127;
  int s = (e < kE) ? src[e] : (e - kE);
  int d = (e < kE) ? dst[e] : (e - kE);
  float norm = rsqrtf(deg[s]) * rsqrtf(deg[d]);   // deg >= 1 due to self loops
  atomicAdd(&agg[(size_t)d * kH + f], h[(size_t)s * kH + f] * norm);
}

// x3 = bn3(elu(agg + b3)) + x1
__global__ void gcn_fin_kernel(const float* __restrict__ agg, const float* __restrict__ bias,
                               const float* __restrict__ g, const float* __restrict__ b,
                               const float* __restrict__ x1, float* __restrict__ x3) {
  int idx = blockIdx.x * blockDim.x + threadIdx.x;
  if (idx >= kN * kH) return;
  int col = idx & 127;
  float v = agg[idx] + bias[col];
  v = (v > 0.f) ? v : (__expf(v) - 1.f);
  x3[idx] = v * (g[col] * rsqrtf(1.f + kEps)) + b[col] + x1[idx];
}

// ------------------------------------------------------------------
// fused edge classifier MLP: one wave handles 16 edge pairs.
// LDS ping-pong: bufA (16x256) <-> bufB (16x128). All GEMMs via fp32 WMMA.
// ------------------------------------------------------------------
__device__ __forceinline__ void mlp_layer_wmma(
    const float* __restrict__ in, int K, const float* __restrict__ W, int OC,
    const float* __restrict__ bias, const float* __restrict__ g, const float* __restrict__ b,
    float* __restrict__ outbuf, int lr, int hi) {
  const float sb = rsqrtf(1.f + kEps);
  for (int ct = 0; ct < OC; ct += 16) {
    v8f acc = {};
    for (int k = 0; k < K; k += 4) {
      const int ka = k + 2 * hi;
      v2f a, wv;
      a.x  = in[lr * K + ka];
      a.y  = in[lr * K + ka + 1];
      wv.x = W[ka * OC + ct + lr];
      wv.y = W[(ka + 1) * OC + ct + lr];
      acc = __builtin_amdgcn_wmma_f32_16x16x4_f32(false, a, false, wv, (short)0, acc,
                                                  false, false);
    }
    const int col = ct + lr;
    const float sc = g[col] * sb;
    const float sh = b[col];
    const float bi = bias[col];
#pragma unroll
    for (int r = 0; r < 8; ++r) {
      float v = (acc[r] + bi) * sc + sh;       // bn(eval) after linear
      v = (v > 0.f) ? v : 0.f;                 // relu
      outbuf[(r + 8 * hi) * OC + col] = v;
    }
  }
}

__global__ void __launch_bounds__(32) mlp_kernel(
    const float* __restrict__ x3, const int* __restrict__ pairs,
    const float* __restrict__ L1W, const float* __restrict__ L1b,
    const float* __restrict__ g1, const float* __restrict__ b1,
    const float* __restrict__ L2W, const float* __restrict__ L2b,
    const float* __restrict__ g2, const float* __restrict__ b2,
    const float* __restrict__ L3W, const float* __restrict__ L3b,
    const float* __restrict__ g3, const float* __restrict__ b3,
    const float* __restrict__ L4W, const float* __restrict__ L4b,
    float* __restrict__ out) {
  __shared__ float bufA[16 * 256];
  __shared__ float bufB[16 * 128];
  const int p0 = blockIdx.x * 16;
  const int l  = threadIdx.x;
  const int lr = l & 15;
  const int hi = l >> 4;

  // gather eemb = [x3[u] ; x3[v]] for 16 pairs -> bufA (16 x 256)
  for (int idx = l; idx < 16 * 256; idx += 32) {
    int row = idx >> 8;
    int col = idx & 255;
    int node = pairs[(p0 + row) * 2 + (col >> 7)];
    bufA[idx] = x3[(size_t)node * kH + (col & 127)];
  }
  __syncthreads();

  mlp_layer_wmma(bufA, 256, L1W, 128, L1b, g1, b1, bufB, lr, hi);
  __syncthreads();
  mlp_layer_wmma(bufB, 128, L2W, 64, L2b, g2, b2, bufA, lr, hi);
  __syncthreads();
  mlp_layer_wmma(bufA, 64, L3W, 32, L3b, g3, b3, bufB, lr, hi);
  __syncthreads();

  // L4: 32 -> 2 (no bn/relu), pad B columns >= 2 with zeros
  {
    v8f acc = {};
    for (int k = 0; k < 32; k += 4) {
      const int ka = k + 2 * hi;
      v2f a, wv;
      a.x  = bufB[lr * 32 + ka];
      a.y  = bufB[lr * 32 + ka + 1];
      wv.x = (lr < 2) ? L4W[ka * 2 + lr] : 0.f;
      wv.y = (lr < 2) ? L4W[(ka + 1) * 2 + lr] : 0.f;
      acc = __builtin_amdgcn_wmma_f32_16x16x4_f32(false, a, false, wv, (short)0, acc,
                                                  false, false);
    }
    if (lr < 2) {
      const float bb = L4b[lr];
#pragma unroll
      for (int r = 0; r < 8; ++r)
        out[(size_t)(p0 + r + 8 * hi) * 2 + lr] = acc[r] + bb;
    }
  }
}

// ------------------------------------------------------------------
// host side
// ------------------------------------------------------------------
static inline int cdiv(int a, int b) { return (a + b - 1) / b; }

static void run_gat(hipStream_t stream, const float* xin, const float* W,
                    const float* av_src, const float* av_dst, const float* bias,
                    const float* bng, const float* bnb, int heads, int ch,
                    const int* src, const int* dst,
                    float* h, float* agg, float* as_, float* ad_,
                    float* emax, float* denom, float* attn, float* xout) {
  dim3 ggrid(kN / 16, kH / 16);
  gemm_wmma_f32_kernel<<<ggrid, 32, 0, stream>>>(xin, W, h, kH, kH);
  const int na = kN * heads;
  alpha_kernel<<<cdiv(na, 256), 256, 0, stream>>>(h, av_src, as_, heads, ch);
  alpha_kernel<<<cdiv(na, 256), 256, 0, stream>>>(h, av_dst, ad_, heads, ch);
  fill_kernel<<<cdiv(na, 256), 256, 0, stream>>>(emax, -3.0e38f, na);
  fill_kernel<<<cdiv(na, 256), 256, 0, stream>>>(denom, 0.f, na);
  fill_kernel<<<cdiv(kN * kH, 256), 256, 0, stream>>>(agg, 0.f, kN * kH);
  const int te = kTotE * heads;
  att_max_kernel<<<cdiv(te, 256), 256, 0, stream>>>(as_, ad_, src, dst, heads, emax);
  att_sum_kernel<<<cdiv(te, 256), 256, 0, stream>>>(as_, ad_, src, dst, heads, emax, denom, attn);
  const int tf = kTotE * kH;
  att_agg_kernel<<<cdiv(tf, 256), 256, 0, stream>>>(h, attn, denom, src, dst, heads, ch, agg);
  bn_elu_kernel<<<cdiv(kN * kH, 256), 256, 0, stream>>>(agg, bias, bng, bnb, xout);
}

extern "C" void kernel_launch(void* const* d_in, const int* in_sizes, int n_in,
                              void* d_out, int out_size, void* d_ws, size_t ws_size,
                              hipStream_t stream) {
  const float* x     = (const float*)d_in[0];
  const int*   ei    = (const int*)d_in[1];
  const int*   src   = ei;
  const int*   dst   = ei + kE;
  const int*   pairs = (const int*)d_in[2];
  const float* W1    = (const float*)d_in[3];
  const float* aS1   = (const float*)d_in[4];
  const float* aD1   = (const float*)d_in[5];
  const float* b1    = (const float*)d_in[6];
  const float* W2    = (const float*)d_in[7];
  const float* aS2   = (const float*)d_in[8];
  const float* aD2   = (const float*)d_in[9];
  const float* b2    = (const float*)d_in[10];
  const float* W3    = (const float*)d_in[11];
  const float* b3    = (const float*)d_in[12];
  const float* bn1g  = (const float*)d_in[13];
  const float* bn1b  = (const float*)d_in[14];
  const float* bn2g  = (const float*)d_in[15];
  const float* bn2b  = (const float*)d_in[16];
  const float* bn3g  = (const float*)d_in[17];
  const float* bn3b  = (const float*)d_in[18];
  const float* L1W   = (const float*)d_in[19];
  const float* L1b   = (const float*)d_in[20];
  const float* mb1g  = (const float*)d_in[21];
  const float* mb1b  = (const float*)d_in[22];
  const float* L2W   = (const float*)d_in[23];
  const float* L2b   = (const float*)d_in[24];
  const float* mb2g  = (const float*)d_in[25];
  const float* mb2b  = (const float*)d_in[26];
  const float* L3W   = (const float*)d_in[27];
  const float* L3b   = (const float*)d_in[28];
  const float* mb3g  = (const float*)d_in[29];
  const float* mb3b  = (const float*)d_in[30];
  const float* L4W   = (const float*)d_in[31];
  const float* L4b   = (const float*)d_in[32];

  float* ws = (float*)d_ws;
  size_t off = 0;
  float* h     = ws + off; off += (size_t)kN * kH;
  float* agg   = ws + off; off += (size_t)kN * kH;
  float* x1    = ws + off; off += (size_t)kN * kH;
  float* x2    = ws + off; off += (size_t)kN * kH;
  float* x3    = ws + off; off += (size_t)kN * kH;
  float* as_   = ws + off; off += (size_t)kN * 4;
  float* ad_   = ws + off; off += (size_t)kN * 4;
  float* emax  = ws + off; off += (size_t)kN * 4;
  float* denom = ws + off; off += (size_t)kN * 4;
  float* attn  = ws + off; off += (size_t)kTotE * 4;
  float* deg   = ws + off; off += (size_t)kN;

  // ---- GAT1: 128 -> 4 heads x 32
  run_gat(stream, x, W1, aS1, aD1, b1, bn1g, bn1b, 4, 32,
          src, dst, h, agg, as_, ad_, emax, denom, attn, x1);
  // ---- GAT2: 128 -> 2 heads x 64
  run_gat(stream, x1, W2, aS2, aD2, b2, bn2g, bn2b, 2, 64,
          src, dst, h, agg, as_, ad_, emax, denom, attn, x2);

  // ---- GCN: 128 -> 128, symmetric norm, then bn(elu(.)) + x1 residual
  fill_kernel<<<cdiv(kN, 256), 256, 0, stream>>>(deg, 0.f, kN);
  deg_kernel<<<cdiv(kTotE, 256), 256, 0, stream>>>(dst, deg);
  dim3 ggrid(kN / 16, kH / 16);
  gemm_wmma_f32_kernel<<<ggrid, 32, 0, stream>>>(x2, W3, h, kH, kH);
  fill_kernel<<<cdiv(kN * kH, 256), 256, 0, stream>>>(agg, 0.f, kN * kH);
  gcn_agg_kernel<<<cdiv(kTotE * kH, 256), 256, 0, stream>>>(h, deg, src, dst, agg);
  gcn_fin_kernel<<<cdiv(kN * kH, 256), 256, 0, stream>>>(agg, b3, bn3g, bn3b, x1, x3);

  // ---- fused edge-pair MLP classifier
  mlp_kernel<<<kM / 16, 32, 0, stream>>>(
      x3, pairs,
      L1W, L1b, mb1g, mb1b,
      L2W, L2b, mb2g, mb2b,
      L3W, L3b, mb3g, mb3b,
      L4W, L4b, (float*)d_out);
}